// GeneralPf_UJUt_32469952758283
// MI455X (gfx1250) — compile-verified
//
#include <hip/hip_runtime.h>
#include <math.h>
#include <stdint.h>

typedef float v2f __attribute__((ext_vector_type(2)));
typedef float v8f __attribute__((ext_vector_type(8)));

#define NFM  2048
#define NOCC 1024
#define NB   16
#define KBLK 16

// ---------------------------------------------------------------------------
// CDNA5 async global->LDS copy (ASYNCcnt-tracked) + explicit wait.
// LDS address is taken from the vdst VGPR; low 32 bits of a generic pointer to a
// __shared__ object are exactly the wave-relative LDS byte offset.
// ---------------------------------------------------------------------------
__device__ __forceinline__ void async_copy_b32(uint32_t lds_addr, const float* gptr) {
  asm volatile("global_load_async_to_lds_b32 %0, %1, off"
               :: "v"(lds_addr), "v"(gptr) : "memory");
}
__device__ __forceinline__ void wait_async0() {
  asm volatile("s_wait_asynccnt 0x0" ::: "memory");
}

// ---------------------------------------------------------------------------
// Js = 0.5*(J - J^T), LDS-tiled so both accesses are coalesced.
// ---------------------------------------------------------------------------
__global__ void skew_kernel(const float* __restrict__ J, float* __restrict__ Js) {
  __shared__ float tile[32][33];
  const int bx = blockIdx.x * 32, by = blockIdx.y * 32;
  const int tx = threadIdx.x;
  for (int ty = threadIdx.y; ty < 32; ty += 8)
    tile[ty][tx] = J[(size_t)(bx + ty) * NFM + by + tx];
  __syncthreads();
  for (int ty = threadIdx.y; ty < 32; ty += 8) {
    float a = J[(size_t)(by + ty) * NFM + bx + tx];
    float b = tile[tx][ty];                 // J[bx+tx][by+ty] == J^T[by+ty][bx+tx]
    Js[(size_t)(by + ty) * NFM + bx + tx] = 0.5f * (a - b);
  }
}

// ---------------------------------------------------------------------------
// Ut = U^T (so both GEMMs below are plain NN).
// ---------------------------------------------------------------------------
__global__ void transpose_kernel(const float* __restrict__ A, float* __restrict__ At) {
  __shared__ float tile[32][33];
  const int bx = blockIdx.x * 32, by = blockIdx.y * 32;
  const int tx = threadIdx.x;
  for (int ty = threadIdx.y; ty < 32; ty += 8)
    tile[ty][tx] = A[(size_t)(by + ty) * NFM + bx + tx];
  __syncthreads();
  for (int ty = threadIdx.y; ty < 32; ty += 8)
    At[(size_t)(bx + ty) * NFM + by + tx] = tile[tx][ty];
}

// ---------------------------------------------------------------------------
// C = A @ B, 2048^3 fp32 via V_WMMA_F32_16X16X4_F32.
// 256 threads = 8 waves (4 M-waves x 2 N-waves), wave tile 32x32, block tile
// 128x64. A (128x16) and B (16x64) K-blocks are double-buffered in LDS via
// async global->LDS copies overlapping the WMMA stream of the previous block.
//
// Fragment layouts per CDNA5 ISA 7.12.2 (wave32):
//   A 16x4 : lane%16 = M row, K = vgpr + 2*(lane>=16)  -> b64 of consecutive K
//   B 4x16 : lane%16 = N col, K = vgpr + 2*(lane>=16)
//   C 16x16: 8 VGPRs, M = vgpr + 8*(lane>=16), N = lane%16
// LDS A padded to stride 18 -> the 16-lane b64 fragment reads hit 16 distinct
// bank pairs (18k mod 64 distinct for k=0..15); LDS B [kk][64] is naturally
// conflict-free for b32 fragment reads.
// ---------------------------------------------------------------------------
__global__ __launch_bounds__(256)
void wmma_gemm_nn(const float* __restrict__ A, const float* __restrict__ B,
                  float* __restrict__ C) {
  __shared__ __align__(16) float ldsA[2][128][18];   // 18 KB
  __shared__ __align__(16) float ldsB[2][KBLK][64];  //  8 KB

  const int t     = threadIdx.x;
  const int lane  = t & 31;
  const int wid   = t >> 5;
  const int waveM = wid & 3;              // 0..3
  const int waveN = wid >> 2;             // 0..1
  const int rowBlk = blockIdx.x * 128;
  const int colBlk = blockIdx.y * 64;
  const int lrow = lane & 15;
  const int half = lane >> 4;             // 0 or 1

  // Coalesced staging: A element e=(r,kk) e=t+256q -> 16-float runs per row;
  // B element e=(kk,c) e=t+256q -> 64-float runs per row.
  const int sAk = t & 15;
  const int sBc = t & 63;

  auto stageA = [&](int kb, int buf) {
    #pragma unroll
    for (int q = 0; q < 8; ++q) {
      const int r = (t >> 4) + 16 * q;
      async_copy_b32((uint32_t)(uintptr_t)&ldsA[buf][r][sAk],
                     A + (size_t)(rowBlk + r) * NFM + kb + sAk);
    }
  };
  auto stageB = [&](int kb, int buf) {
    #pragma unroll
    for (int q = 0; q < 4; ++q) {
      const int kk = (t >> 6) + 4 * q;
      async_copy_b32((uint32_t)(uintptr_t)&ldsB[buf][kk][sBc],
                     B + (size_t)(kb + kk) * NFM + colBlk + sBc);
    }
  };

  v8f acc00 = {}, acc01 = {}, acc10 = {}, acc11 = {};

  stageA(0, 0);
  stageB(0, 0);
  int cur = 0;
  const int r0 = waveM * 32 + lrow;
  const int c0 = waveN * 32 + lrow;

  for (int kb = 0; kb < NFM; kb += KBLK) {
    wait_async0();        // my async group for buf[cur] is complete
    __syncthreads();      // => everyone's group for buf[cur] is complete,
                          //    and nobody still reads buf[cur^1]
    if (kb + KBLK < NFM) {
      stageA(kb + KBLK, cur ^ 1);   // overlaps with WMMA stream below
      stageB(kb + KBLK, cur ^ 1);
    }
    #pragma unroll
    for (int s = 0; s < 4; ++s) {
      const int kk = 4 * s + 2 * half;
      v2f a0 = *(const v2f*)&ldsA[cur][r0][kk];        // K = kk, kk+1
      v2f a1 = *(const v2f*)&ldsA[cur][r0 + 16][kk];
      v2f b0, b1;
      b0.x = ldsB[cur][kk][c0];        b0.y = ldsB[cur][kk + 1][c0];
      b1.x = ldsB[cur][kk][c0 + 16];   b1.y = ldsB[cur][kk + 1][c0 + 16];
      acc00 = __builtin_amdgcn_wmma_f32_16x16x4_f32(false, a0, false, b0, (short)0, acc00, false, false);
      acc01 = __builtin_amdgcn_wmma_f32_16x16x4_f32(false, a0, false, b1, (short)0, acc01, false, false);
      acc10 = __builtin_amdgcn_wmma_f32_16x16x4_f32(false, a1, false, b0, (short)0, acc10, false, false);
      acc11 = __builtin_amdgcn_wmma_f32_16x16x4_f32(false, a1, false, b1, (short)0, acc11, false, false);
    }
    cur ^= 1;
  }

  const int mOff = 8 * half;
  v8f accs[2][2] = {{acc00, acc01}, {acc10, acc11}};
  #pragma unroll
  for (int tm = 0; tm < 2; ++tm)
    #pragma unroll
    for (int tn = 0; tn < 2; ++tn) {
      float* pc = C + (size_t)(rowBlk + waveM * 32 + tm * 16 + mOff) * NFM
                    + colBlk + waveN * 32 + tn * 16 + lrow;
      #pragma unroll
      for (int v = 0; v < 8; ++v)
        pc[(size_t)v * NFM] = accs[tm][tn][v];
    }
}

// ---------------------------------------------------------------------------
// sub[b][i][j] = F[idx[b][i]][idx[b][j]]
// ---------------------------------------------------------------------------
__global__ void gather_kernel(const float* __restrict__ F, const int* __restrict__ idx,
                              float* __restrict__ sub) {
  const int b = blockIdx.y;
  const int t = blockIdx.x * 256 + threadIdx.x;   // 0 .. NOCC*NOCC-1
  const int i = t >> 10, j = t & 1023;
  const int ri = idx[b * NOCC + i];
  const int cj = idx[b * NOCC + j];
  sub[(size_t)b * NOCC * NOCC + (size_t)i * NOCC + j] = F[(size_t)ri * NFM + cj];
}

// ---------------------------------------------------------------------------
// Parlett-Reid Pfaffian with partial pivoting. One 1024-thread workgroup per
// batch matrix; matrix lives in global memory (4 MB >> LDS), tau/cn + argmax
// reduction in LDS. Only the trailing (k+2:,k+2:) block is updated — tau and
// cn are zero for indices <= k+1, so this is exactly the reference update.
// ---------------------------------------------------------------------------
__global__ __launch_bounds__(1024)
void pfaffian_kernel(float* __restrict__ subAll, float* __restrict__ out) {
  const int n = NOCC;
  const int b = blockIdx.x;
  float* A = subAll + (size_t)b * n * n;
  const int tid = threadIdx.x;

  __shared__ float sval[1024];
  __shared__ int   sidx[1024];
  __shared__ float tau[1024];
  __shared__ float cn[1024];
  __shared__ float s_sign, s_log;

  if (tid == 0) { s_sign = 1.f; s_log = 0.f; }

  for (int k = 0; k < n; k += 2) {
    // ---- pivot search: argmax_{i>k} |A[i,k]|, first index on ties ----
    sval[tid] = (tid > k) ? fabsf(A[(size_t)tid * n + k]) : -__builtin_inff();
    sidx[tid] = tid;
    __syncthreads();
    for (int s = 512; s > 0; s >>= 1) {
      if (tid < s) {
        float v2 = sval[tid + s]; int i2 = sidx[tid + s];
        if (v2 > sval[tid] || (v2 == sval[tid] && i2 < sidx[tid])) {
          sval[tid] = v2; sidx[tid] = i2;
        }
      }
      __syncthreads();
    }
    const int kp = sidx[0];

    // ---- swap row k+1 <-> kp, then column k+1 <-> kp ----
    if (kp != k + 1) {
      float r1 = A[(size_t)(k + 1) * n + tid];
      float r2 = A[(size_t)kp * n + tid];
      A[(size_t)(k + 1) * n + tid] = r2;
      A[(size_t)kp * n + tid]      = r1;
      __syncthreads();
      float c1 = A[(size_t)tid * n + (k + 1)];
      float c2 = A[(size_t)tid * n + kp];
      A[(size_t)tid * n + (k + 1)] = c2;
      A[(size_t)tid * n + kp]      = c1;
      __syncthreads();
    }

    const float pivot = A[(size_t)k * n + (k + 1)];

    // ---- tau = A[:,k]/pivot, cn = A[:,k+1] (masked i > k+1) ----
    tau[tid] = (tid > k + 1) ? A[(size_t)tid * n + k] / pivot : 0.f;
    cn[tid]  = (tid > k + 1) ? A[(size_t)tid * n + (k + 1)]    : 0.f;
    __syncthreads();

    // ---- rank-2 update of trailing block: A += tau cn^T - cn tau^T ----
    const int lane = tid & 31, warp = tid >> 5;
    for (int i = k + 2 + warp; i < n; i += 32) {
      const float ti = tau[i], ci = cn[i];
      float* Ai = A + (size_t)i * n;
      for (int j = k + 2 + lane; j < n; j += 32)
        Ai[j] += ti * cn[j] - ci * tau[j];
    }

    if (tid == 0) {
      float sgn = (pivot > 0.f) ? 1.f : ((pivot < 0.f) ? -1.f : 0.f);
      if (kp != k + 1) s_sign = -s_sign;
      s_sign *= sgn;
      s_log  += logf(fabsf(pivot));
    }
    __syncthreads();   // updates + sign/log visible before next iteration
  }
  if (tid == 0) { out[b] = s_sign; out[NB + b] = s_log; }
}

// ---------------------------------------------------------------------------
// Workspace layout (floats), total 20M floats = 80 MB:
//   [0,    N2 ) : Js, later overwritten by F
//   [N2,  2N2 ) : Ut, later overwritten by sub
//   [2N2, 3N2 ) : T = Js @ Ut, later overwritten by sub
//   [N2, N2+B*M*M) : sub (64 MB), written after Ut/T are dead
// ---------------------------------------------------------------------------
extern "C" void kernel_launch(void* const* d_in, const int* in_sizes, int n_in,
                              void* d_out, int out_size, void* d_ws, size_t ws_size,
                              hipStream_t stream) {
  (void)in_sizes; (void)n_in; (void)out_size; (void)ws_size;
  const float* U   = (const float*)d_in[0];
  const float* J   = (const float*)d_in[1];
  const int*   idx = (const int*)d_in[2];
  float* out = (float*)d_out;
  float* ws  = (float*)d_ws;

  const size_t N2 = (size_t)NFM * NFM;
  float* Js  = ws;            // dead after T
  float* Ut  = ws + N2;       // dead after T
  float* T   = ws + 2 * N2;   // dead after F
  float* F   = ws;            // overwrites Js
  float* sub = ws + N2;       // overwrites Ut, T

  dim3 tb(32, 8);
  dim3 tg(NFM / 32, NFM / 32);
  skew_kernel     <<<tg, tb, 0, stream>>>(J, Js);
  transpose_kernel<<<tg, tb, 0, stream>>>(U, Ut);

  dim3 gg(NFM / 128, NFM / 64);
  wmma_gemm_nn<<<gg, 256, 0, stream>>>(Js, Ut, T);   // T = Js @ U^T
  wmma_gemm_nn<<<gg, 256, 0, stream>>>(U,  T,  F);   // F = U @ T

  dim3 gatherg((NOCC * NOCC) / 256, NB);
  gather_kernel<<<gatherg, 256, 0, stream>>>(F, idx, sub);

  pfaffian_kernel<<<NB, 1024, 0, stream>>>(sub, out);
}